// Model_3582002725484
// MI455X (gfx1250) — compile-verified
//
#include <hip/hip_runtime.h>

typedef __attribute__((ext_vector_type(2))) float v2f;
typedef __attribute__((ext_vector_type(8))) float v8f;

#define S     8192
#define EPS   1e-5f
#define TILE  2048      // k/v positions staged in LDS per pass (32 KB as float4)
#define QPB   64        // queries per block (4 threads per query, 256 threads)
#define BSTR  132       // padded LDS row stride for B tile (banks 4n+j, conflict-free)

// ---------------------------------------------------------------------------
// Kernel 1: h = relu((x .* mask) @ inn_w^T + inn_b)   [8192,128]x[128,2]
// fp32 WMMA 16x16x4: one wave per 16-row tile, N padded 2->16.
// B^T staged zero-padded in LDS once (no divergent loads in the MMA loop).
// A layout (16x4 f32): lanes 0-15 = rows, VGPR0/1 = K{0,1}; lanes 16-31 = K{2,3}.
// C/D: VGPR r -> M = r + 8*(lane>>4), N = lane&15.
// ---------------------------------------------------------------------------
__global__ __launch_bounds__(256) void k_input_gemm(
    const float* __restrict__ x, const float* __restrict__ mask,
    const float* __restrict__ w, const float* __restrict__ b,
    float* __restrict__ h)
{
  __shared__ float blds[16 * BSTR];   // padded B: [n][k], n>=2 rows are zero

  const int tid = threadIdx.x;
  for (int i = tid; i < 16 * BSTR; i += 256) blds[i] = 0.0f;
  __syncthreads();
  if (tid < 256) {                    // 256 = 2 rows x 128 cols of inn_w
    const int nn = tid >> 7, kk = tid & 127;
    blds[nn * BSTR + kk] = w[tid];
  }
  __syncthreads();

  const int lane = threadIdx.x & 31;
  const int wave = threadIdx.x >> 5;
  const int tile = blockIdx.x * 8 + wave;
  const int row  = tile * 16 + (lane & 15);
  const int kh   = (lane >> 4) * 2;           // K sub-offset: 0 or 2
  const int n    = lane & 15;

  const float* xr = x    + row * 128;
  const float* mr = mask + row * 128;
  const float* bl = blds + n * BSTR + kh;

  v8f c = {};
#pragma unroll 4
  for (int k0 = 0; k0 < 128; k0 += 4) {
    v2f A, B;
    A.x = xr[k0 + kh]     * mr[k0 + kh];
    A.y = xr[k0 + kh + 1] * mr[k0 + kh + 1];
    B.x = bl[k0];
    B.y = bl[k0 + 1];
    c = __builtin_amdgcn_wmma_f32_16x16x4_f32(false, A, false, B,
                                              (short)0, c, false, false);
  }

  if (n < 2) {
    const float bias  = b[n];
    const int   rbase = tile * 16 + 8 * (lane >> 4);
#pragma unroll
    for (int r = 0; r < 8; ++r)
      h[(rbase + r) * 2 + n] = fmaxf(c[r] + bias, 0.0f);
  }
}

// ---------------------------------------------------------------------------
// Fused layer: MHA(dh=1, rank-1 scores, flash-style) + residual + LN
//   mode bit0: + FFN + residual + LN2
//   mode bit1: + final LayerNorm (fns/fnb)
//   mode bit2: + tanh(z @ finw^T + finb) -> out[S] ; else out[S,2]
// ---------------------------------------------------------------------------
__global__ __launch_bounds__(256) void k_attn_layer(
    const float* __restrict__ qsrc, const float* __restrict__ kvsrc,
    const float* __restrict__ qkv_w, const float* __restrict__ qkv_b,
    const float* __restrict__ ow,   const float* __restrict__ ob,
    const float* __restrict__ ln1s, const float* __restrict__ ln1b,
    const float* __restrict__ f1w,  const float* __restrict__ f1b,
    const float* __restrict__ f2w,  const float* __restrict__ f2b,
    const float* __restrict__ ln2s, const float* __restrict__ ln2b,
    const float* __restrict__ fns,  const float* __restrict__ fnb,
    const float* __restrict__ finw, const float* __restrict__ finb,
    float* __restrict__ out, int mode)
{
  __shared__ float4 skv[TILE];   // per position: (k0, v0, k1, v1)

  const int tid = threadIdx.x;
  const int q   = blockIdx.x * QPB + (tid >> 2);
  const int r   = tid & 3;                      // j-split lane within query group

  // packed in_proj: rows 0-1 = Wq, 2-3 = Wk, 4-5 = Wv (row-major [6][2])
  const float wq00 = qkv_w[0],  wq01 = qkv_w[1];
  const float wq10 = qkv_w[2],  wq11 = qkv_w[3];
  const float wk00 = qkv_w[4],  wk01 = qkv_w[5];
  const float wk10 = qkv_w[6],  wk11 = qkv_w[7];
  const float wv00 = qkv_w[8],  wv01 = qkv_w[9];
  const float wv10 = qkv_w[10], wv11 = qkv_w[11];
  const float bq0 = qkv_b[0], bq1 = qkv_b[1];
  const float bk0 = qkv_b[2], bk1 = qkv_b[3];
  const float bv0 = qkv_b[4], bv1 = qkv_b[5];

  const float s0 = qsrc[q * 2], s1 = qsrc[q * 2 + 1];
  const float LOG2E = 1.442695040888963f;
  const float q0l = (s0 * wq00 + s1 * wq01 + bq0) * LOG2E;  // head0 query (log2 dom)
  const float q1l = (s0 * wq10 + s1 * wq11 + bq1) * LOG2E;  // head1 query

  float e0 = 0.f, ev0 = 0.f, e1 = 0.f, ev1 = 0.f;

  for (int tb = 0; tb < S; tb += TILE) {
    for (int i = tid; i < TILE; i += 256) {
      const int j = tb + i;
      const float a = kvsrc[j * 2], bb = kvsrc[j * 2 + 1];
      skv[i] = make_float4(a * wk00 + bb * wk01 + bk0,
                           a * wv00 + bb * wv01 + bv0,
                           a * wk10 + bb * wk11 + bk1,
                           a * wv10 + bb * wv11 + bv1);
    }
    __syncthreads();
    // thread r handles j == r (mod 4): 4 distinct 16B addresses/wave, broadcast
    for (int t = 0; t < TILE / 4; ++t) {
      const float4 kv = skv[(t << 2) + r];
      const float e = __builtin_amdgcn_exp2f(q0l * kv.x);  e0 += e;  ev0 += e * kv.y;
      const float f = __builtin_amdgcn_exp2f(q1l * kv.z);  e1 += f;  ev1 += f * kv.w;
    }
    __syncthreads();
  }

  // reduce the 4 partial sums per query (lanes 4q..4q+3 within a wave32)
  for (int msk = 1; msk < 4; msk <<= 1) {
    e0  += __shfl_xor(e0,  msk, 32);
    ev0 += __shfl_xor(ev0, msk, 32);
    e1  += __shfl_xor(e1,  msk, 32);
    ev1 += __shfl_xor(ev1, msk, 32);
  }

  if (r == 0) {
    const float o0 = ev0 / e0;                     // softmax-weighted v, head0
    const float o1 = ev1 / e1;                     // head1
    const float a0 = o0 * ow[0] + o1 * ow[1] + ob[0];
    const float a1 = o0 * ow[2] + o1 * ow[3] + ob[1];

    float y0 = s0 + a0, y1 = s1 + a1;              // residual
    float mn  = 0.5f * (y0 + y1);                  // LN1 (D=2 closed form)
    float d0  = y0 - mn, d1 = y1 - mn;
    float inv = rsqrtf(0.5f * (d0 * d0 + d1 * d1) + EPS);
    float z0  = d0 * inv * ln1s[0] + ln1b[0];
    float z1  = d1 * inv * ln1s[1] + ln1b[1];

    if (mode & 1) {                                // FFN (FF=3) + residual + LN2
      float acc0 = f2b[0], acc1 = f2b[1];
#pragma unroll
      for (int f = 0; f < 3; ++f) {
        const float zf = fmaxf(z0 * f1w[f * 2] + z1 * f1w[f * 2 + 1] + f1b[f], 0.0f);
        acc0 += zf * f2w[f];
        acc1 += zf * f2w[3 + f];
      }
      y0 = z0 + acc0;  y1 = z1 + acc1;
      mn  = 0.5f * (y0 + y1);
      d0  = y0 - mn;  d1 = y1 - mn;
      inv = rsqrtf(0.5f * (d0 * d0 + d1 * d1) + EPS);
      z0  = d0 * inv * ln2s[0] + ln2b[0];
      z1  = d1 * inv * ln2s[1] + ln2b[1];
    }
    if (mode & 2) {                                // final LayerNorm
      mn  = 0.5f * (z0 + z1);
      d0  = z0 - mn;  d1 = z1 - mn;
      inv = rsqrtf(0.5f * (d0 * d0 + d1 * d1) + EPS);
      z0  = d0 * inv * fns[0] + fnb[0];
      z1  = d1 * inv * fns[1] + fnb[1];
    }
    if (mode & 4) {                                // output head -> [S,1]
      out[q] = tanhf(z0 * finw[0] + z1 * finw[1] + finb[0]);
    } else {
      out[q * 2]     = z0;
      out[q * 2 + 1] = z1;
    }
  }
}

// ---------------------------------------------------------------------------
extern "C" void kernel_launch(void* const* d_in, const int* in_sizes, int n_in,
                              void* d_out, int out_size, void* d_ws, size_t ws_size,
                              hipStream_t stream)
{
  const float* x      = (const float*)d_in[0];
  const float* mask   = (const float*)d_in[1];
  const float* inn_w  = (const float*)d_in[2];
  const float* inn_b  = (const float*)d_in[3];
  const float* eqkvw  = (const float*)d_in[4];   // [2,6,2]
  const float* eqkvb  = (const float*)d_in[5];   // [2,6]
  const float* eow    = (const float*)d_in[6];   // [2,2,2]
  const float* eob    = (const float*)d_in[7];   // [2,2]
  const float* eln1s  = (const float*)d_in[8];
  const float* eln1b  = (const float*)d_in[9];
  const float* eln2s  = (const float*)d_in[10];
  const float* eln2b  = (const float*)d_in[11];
  const float* ef1w   = (const float*)d_in[12];  // [2,3,2]
  const float* ef1b   = (const float*)d_in[13];  // [2,3]
  const float* ef2w   = (const float*)d_in[14];  // [2,2,3]
  const float* ef2b   = (const float*)d_in[15];  // [2,2]
  const float* enorms = (const float*)d_in[16];
  const float* enormb = (const float*)d_in[17];
  const float* saqkvw = (const float*)d_in[18];
  const float* saqkvb = (const float*)d_in[19];
  const float* saow   = (const float*)d_in[20];
  const float* saob   = (const float*)d_in[21];
  const float* caqkvw = (const float*)d_in[22];
  const float* caqkvb = (const float*)d_in[23];
  const float* caow   = (const float*)d_in[24];
  const float* caob   = (const float*)d_in[25];
  const float* dln1s  = (const float*)d_in[26];
  const float* dln1b  = (const float*)d_in[27];
  const float* dln2s  = (const float*)d_in[28];
  const float* dln2b  = (const float*)d_in[29];
  const float* dln3s  = (const float*)d_in[30];
  const float* dln3b  = (const float*)d_in[31];
  const float* dff1w  = (const float*)d_in[32];
  const float* dff1b  = (const float*)d_in[33];
  const float* dff2w  = (const float*)d_in[34];
  const float* dff2b  = (const float*)d_in[35];
  const float* dnorms = (const float*)d_in[36];
  const float* dnormb = (const float*)d_in[37];
  const float* outw   = (const float*)d_in[38];
  const float* outb   = (const float*)d_in[39];

  float* ws  = (float*)d_ws;
  float* h   = ws;            // [S,2] input projection (decoder tgt too)
  float* buf = ws + 2 * S;    // [S,2] encoder ping
  float* mem = ws + 4 * S;    // [S,2] encoder memory
  float* t1  = ws + 6 * S;    // [S,2] decoder after self-attn

  const int ablocks = S / QPB;   // 128

  // 1) input GEMM via fp32 WMMA
  k_input_gemm<<<S / 128, 256, 0, stream>>>(x, mask, inn_w, inn_b, h);

  // 2) encoder layer 0: attn + LN1 + FFN + LN2
  k_attn_layer<<<ablocks, 256, 0, stream>>>(h, h,
      eqkvw, eqkvb, eow, eob, eln1s, eln1b,
      ef1w, ef1b, ef2w, ef2b, eln2s, eln2b,
      enorms, enormb, outw, outb, buf, /*mode=*/1);

  // 3) encoder layer 1 (+ final encoder norm -> mem)
  k_attn_layer<<<ablocks, 256, 0, stream>>>(buf, buf,
      eqkvw + 12, eqkvb + 6, eow + 4, eob + 2, eln1s + 2, eln1b + 2,
      ef1w + 6, ef1b + 3, ef2w + 6, ef2b + 2, eln2s + 2, eln2b + 2,
      enorms, enormb, outw, outb, mem, /*mode=*/3);

  // 4) decoder self-attn: attn + LN1 only
  k_attn_layer<<<ablocks, 256, 0, stream>>>(h, h,
      saqkvw, saqkvb, saow, saob, dln1s, dln1b,
      dff1w, dff1b, dff2w, dff2b, dln3s, dln3b,
      dnorms, dnormb, outw, outb, t1, /*mode=*/0);

  // 5) decoder cross-attn + LN2 + FFN + LN3 + dec_norm + tanh head -> d_out[S]
  k_attn_layer<<<ablocks, 256, 0, stream>>>(t1, mem,
      caqkvw, caqkvb, caow, caob, dln2s, dln2b,
      dff1w, dff1b, dff2w, dff2b, dln3s, dln3b,
      dnorms, dnormb, outw, outb, (float*)d_out, /*mode=*/7);
}